// GraphConvolution_1984274891520
// MI455X (gfx1250) — compile-verified
//
#include <hip/hip_runtime.h>
#include <hip/hip_bf16.h>

// ---------------------------------------------------------------------------
// GCNII layer:  out = theta * (S @ W) + (1-theta) * S
//               S   = (1-alpha) * spmm(adj, x) + alpha * h0
// N=50000, E=600000, D=128, alpha=0.1, theta=log(1.5)
// ---------------------------------------------------------------------------

#define DDIM 128
#define ALPHA_C 0.1f
#define THETA_C 0.40546510810816444f  // log(1.5)

typedef __attribute__((ext_vector_type(2))) float v2f;
typedef __attribute__((ext_vector_type(8))) float v8f;

__device__ __forceinline__ void atomic_add_f32(float* p, float v) {
    // hardware global_atomic_add_f32 (no return) — hi lives in L2 (25.6MB << 192MB)
    unsafeAtomicAdd(p, v);
}

// ---- zero the accumulator buffer (float4 stores) ---------------------------
__global__ void gcn_zero_kernel(float4* __restrict__ p, int n4) {
    int i = blockIdx.x * blockDim.x + threadIdx.x;
    if (i < n4) p[i] = make_float4(0.f, 0.f, 0.f, 0.f);
}

// ---- SpMM scatter: one wave32 per edge, 4 floats per lane ------------------
__global__ void gcn_spmm_kernel(const float* __restrict__ x,
                                const float* __restrict__ vals,
                                const int*   __restrict__ rows,
                                const int*   __restrict__ cols,
                                float*       __restrict__ hi,
                                int nEdges) {
    int gid   = blockIdx.x * blockDim.x + threadIdx.x;
    int edge  = gid >> 5;            // 32 lanes share one edge (wave-uniform)
    int chunk = (gid & 31) * 4;      // 4 consecutive floats per lane
    if (edge >= nEdges) return;
    float v = vals[edge];
    int r = rows[edge];
    int c = cols[edge];
    const float4 xv = *(const float4*)(x + (size_t)c * DDIM + chunk);
    float* dst = hi + (size_t)r * DDIM + chunk;
    atomic_add_f32(dst + 0, v * xv.x);
    atomic_add_f32(dst + 1, v * xv.y);
    atomic_add_f32(dst + 2, v * xv.z);
    atomic_add_f32(dst + 3, v * xv.w);
}

// ---- support: S = (1-alpha)*hi + alpha*h0, in place over hi ----------------
__global__ void gcn_support_kernel(float4* __restrict__ hi,
                                   const float4* __restrict__ h0, int n4) {
    int i = blockIdx.x * blockDim.x + threadIdx.x;
    if (i < n4) {
        float4 a = hi[i];
        float4 b = h0[i];
        float4 o;
        o.x = (1.0f - ALPHA_C) * a.x + ALPHA_C * b.x;
        o.y = (1.0f - ALPHA_C) * a.y + ALPHA_C * b.y;
        o.z = (1.0f - ALPHA_C) * a.z + ALPHA_C * b.z;
        o.w = (1.0f - ALPHA_C) * a.w + ALPHA_C * b.w;
        hi[i] = o;
    }
}

// ---- fused GEMM + residual with V_WMMA_F32_16X16X4_F32 ---------------------
// Block = 256 threads = 8 waves. Block stages W (128x128 f32 = 64KB) in LDS.
// Wave w computes the 16x128 output tile for rows [blk*128 + w*16, +16).
__global__ __launch_bounds__(256) void gcn_gemm_wmma_kernel(
        const float* __restrict__ S,       // support, [N, 128]
        const float* __restrict__ W,       // weight,  [128, 128] row-major
        float*       __restrict__ out,     // [N, 128]
        int nRows) {
    __shared__ float Wlds[DDIM * DDIM];    // 64KB, row-major W[k][n]

    const int tid = threadIdx.x;
    // cooperative, coalesced copy of W into LDS
    for (int t = tid; t < DDIM * DDIM; t += 256) {
        Wlds[t] = W[t];
    }
    __syncthreads();

    const int wave = tid >> 5;
    const int lane = tid & 31;
    const int row0 = blockIdx.x * 128 + wave * 16;

    // A-fragment layout (16x4 f32): lane l -> M = l%16, K = 2*(l/16)+{0,1}
    const int m    = lane & 15;
    const int kh   = (lane >> 4) << 1;     // 0 or 2
    int rowA = row0 + m;
    if (rowA >= nRows) rowA = nRows - 1;   // clamp loads; stores guarded later
    const float* Arow = S + (size_t)rowA * DDIM;

    v8f acc[8] = {};                        // 8 column tiles of 16

    for (int k0 = 0; k0 < DDIM; k0 += 4) {
        const int kk = k0 + kh;
        const v2f a = *(const v2f*)(Arow + kk);      // global_load_b64
#pragma unroll
        for (int j = 0; j < 8; ++j) {
            const int n0 = j * 16;
            v2f b;
            // B-fragment (4x16 f32): lane l -> N = l%16, K = 2*(l/16)+{0,1}
            b.x = Wlds[(kk + 0) * DDIM + n0 + m];
            b.y = Wlds[(kk + 1) * DDIM + n0 + m];
            acc[j] = __builtin_amdgcn_wmma_f32_16x16x4_f32(
                false, a, false, b, (short)0, acc[j], false, false);
        }
    }

    // Epilogue: out = theta*acc + (1-theta)*S ; C/D layout:
    // VGPR r, lane l -> M = r + 8*(l/16), N = l%16
    const int half = lane >> 4;
    const int n    = lane & 15;
#pragma unroll
    for (int j = 0; j < 8; ++j) {
        const int col = j * 16 + n;
#pragma unroll
        for (int r = 0; r < 8; ++r) {
            const int row = row0 + r + 8 * half;
            if (row < nRows) {
                const float s = S[(size_t)row * DDIM + col];
                out[(size_t)row * DDIM + col] =
                    THETA_C * acc[j][r] + (1.0f - THETA_C) * s;
            }
        }
    }
}

// ---------------------------------------------------------------------------
extern "C" void kernel_launch(void* const* d_in, const int* in_sizes, int n_in,
                              void* d_out, int out_size, void* d_ws, size_t ws_size,
                              hipStream_t stream) {
    const float* x        = (const float*)d_in[0];
    const float* h0       = (const float*)d_in[1];
    const float* adj_vals = (const float*)d_in[2];
    const float* weight   = (const float*)d_in[3];
    const int*   adj_rows = (const int*)d_in[4];
    const int*   adj_cols = (const int*)d_in[5];
    float*       out      = (float*)d_out;

    const int nNodes = in_sizes[0] / DDIM;   // 50000
    const int nEdges = in_sizes[2];          // 600000
    const int total  = nNodes * DDIM;        // 6.4M floats
    const int n4     = total / 4;

    float* hi = (float*)d_ws;                // 25.6 MB scratch: hi, then S in place

    // 1) hi = 0
    gcn_zero_kernel<<<(n4 + 255) / 256, 256, 0, stream>>>((float4*)hi, n4);

    // 2) hi += scatter(adj_vals * x[cols]) ; one wave per edge
    {
        long threads = (long)nEdges * 32;
        int  blocks  = (int)((threads + 255) / 256);
        gcn_spmm_kernel<<<blocks, 256, 0, stream>>>(x, adj_vals, adj_rows,
                                                    adj_cols, hi, nEdges);
    }

    // 3) S = 0.9*hi + 0.1*h0  (in place)
    gcn_support_kernel<<<(n4 + 255) / 256, 256, 0, stream>>>(
        (float4*)hi, (const float4*)h0, n4);

    // 4) out = theta*(S @ W) + (1-theta)*S   (WMMA f32 16x16x4)
    {
        int blocks = (nNodes + 127) / 128;
        gcn_gemm_wmma_kernel<<<blocks, 256, 0, stream>>>(hi, weight, out, nNodes);
    }
}